// CTCLoss_79542794322248
// MI455X (gfx1250) — compile-verified
//
#include <hip/hip_runtime.h>
#include <hip/hip_bf16.h>
#include <stdint.h>

#define T_DIM 512
#define B_DIM 32
#define V_DIM 4096
#define LMAX  32
#define NEGV  (-1e30f)

typedef unsigned int u32x4 __attribute__((ext_vector_type(4)));
typedef int          i32x8 __attribute__((ext_vector_type(8)));
typedef float        v2f   __attribute__((ext_vector_type(2)));
typedef float        v8f   __attribute__((ext_vector_type(8)));

// ---------------------------------------------------------------------------
// Kernel 1: per-row logsumexp over V. One wave32 per (t,b) row.
// The 16KB row is DMA'd into LDS by the Tensor Data Mover (inline asm,
// VIMAGE 2-group form: groups 0+1 in SGPRs, VADDR2/3 = NULL for 2D),
// then the wave does two passes (max, sum-exp) over LDS.
// ---------------------------------------------------------------------------
__global__ void __launch_bounds__(32) lse_kernel(const float* __restrict__ acts,
                                                 float* __restrict__ lse) {
  const int row  = blockIdx.x;            // row = t*B + b, V_DIM floats
  const int lane = threadIdx.x;           // 0..31 (wave32)
  const float* __restrict__ rowp = acts + (size_t)row * V_DIM;

  __shared__ float4 buf4[V_DIM / 4];      // 16 KB, offset 0 in LDS

  // ---- Tensor DMA descriptor (D#), per CDNA5 ISA ----
  // 2D tensor == tile: 1 row x 4096 f32 elements, destination LDS offset 0.
  const uint64_t ga = (uint64_t)(uintptr_t)rowp;
  u32x4 g0;
  g0[0] = 1u;                                       // count=1, user mode
  g0[1] = 0u;                                       // lds_addr = 0 (buf4)
  g0[2] = (unsigned)(ga & 0xFFFFFFFFu);             // global_addr[31:0]
  g0[3] = (unsigned)(((ga >> 32) & 0x01FFFFFFu)     // global_addr[56:32]
                     | (2u << 30));                 // type = 2 ("image")
  i32x8 g1;
  g1[0] = 0x00020000;                 // data_size=2 (4 bytes); mask/flags = 0
  g1[1] = (int)(4096u << 16);         // tensor_dim0 = 4096 (low16 @ bits63:48)
  g1[2] = (int)(1u << 16);            // tensor_dim0 hi=0; tensor_dim1 = 1
  g1[3] = (int)(4096u << 16);         // tensor_dim1 hi=0; tile_dim0 = 4096
  g1[4] = 1;                          // tile_dim1 = 1; tile_dim2 = 0 (unused)
  g1[5] = 4096;                       // tensor_dim0_stride low32
  g1[6] = (int)(4096u << 16);         // dim0_stride hi=0; dim1_stride low16
  g1[7] = 0;                          // dim1_stride hi

  // Issue the TDM load (one instruction per wave; EXEC ignored).
  asm volatile("tensor_load_to_lds %0, %1"
               :
               : "s"(g0), "s"(g1)
               : "memory");

#if __has_builtin(__builtin_amdgcn_s_wait_tensorcnt)
  __builtin_amdgcn_s_wait_tensorcnt(0);
#else
  asm volatile("s_wait_tensorcnt 0x0" ::: "memory");
#endif

  // Pass 1: row max (128 elems/lane, conflict-free float4 reads).
  float M = -INFINITY;
  #pragma unroll
  for (int i = 0; i < 32; ++i) {
    float4 x = buf4[i * 32 + lane];
    M = fmaxf(M, fmaxf(fmaxf(x.x, x.y), fmaxf(x.z, x.w)));
  }
  #pragma unroll
  for (int m = 16; m; m >>= 1) M = fmaxf(M, __shfl_xor(M, m, 32));

  // Pass 2: sum of exp(x - M).
  float S = 0.0f;
  #pragma unroll
  for (int i = 0; i < 32; ++i) {
    float4 x = buf4[i * 32 + lane];
    S += expf(x.x - M) + expf(x.y - M) + expf(x.z - M) + expf(x.w - M);
  }
  #pragma unroll
  for (int m = 16; m; m >>= 1) S += __shfl_xor(S, m, 32);

  if (lane == 0) lse[row] = M + logf(S);
}

// ---------------------------------------------------------------------------
// Kernel 2: CTC alpha recursion. One wave32 per batch element.
// Lane l owns states s=2l and s=2l+1; lane 31 additionally owns s=64.
// Neighbor alphas via one __shfl_up per step; no barriers, no LDS.
// ---------------------------------------------------------------------------
__device__ __forceinline__ float lae2(float a, float b) {
  float m = fmaxf(a, b);
  return m + logf(expf(a - m) + expf(b - m));
}
__device__ __forceinline__ float lae3(float a, float b, float c) {
  float m = fmaxf(a, fmaxf(b, c));
  return m + logf(expf(a - m) + expf(b - m) + expf(c - m));
}

__global__ void __launch_bounds__(32) ctc_alpha_kernel(
    const float* __restrict__ acts, const int* __restrict__ labels,
    const int* __restrict__ act_lens, const int* __restrict__ label_lens,
    const float* __restrict__ lse, float* __restrict__ loss_out) {
  const int b    = blockIdx.x;
  const int lane = threadIdx.x;

  // Offset of this example's labels in the flat concat.
  int off = 0;
  for (int i = 0; i < b; ++i) off += label_lens[i];
  const int L  = label_lens[b];           // 16..32
  const int Sb = 2 * L + 1;               // valid lattice states
  const int Tb = act_lens[b];             // <= T

  const int s0 = 2 * lane;                // even (blank) state
  const int s1 = 2 * lane + 1;            // odd (label) state
  const int e1 = (lane < L) ? labels[off + lane] : 0;   // ext[s1]
  const int e1prev = __shfl_up(e1, 1, 32);              // ext[s1-2]
  const bool cs1 = (s1 >= 2) && (e1 != 0) && (e1 != e1prev);
  const bool valid0 = (s0 < Sb);
  const bool valid1 = (s1 < Sb);
  const bool valid64 = (64 < Sb);         // only when L == 32

  const float* __restrict__ actb = acts + (size_t)b * V_DIM;
  const size_t tstride = (size_t)B_DIM * V_DIM;

  // t = 0 init.
  float blk = actb[0];
  float lab = actb[e1];
  float l0  = lse[b];
  float A0 = (lane == 0) ? (blk - l0) : NEGV;   // s=0
  float A1 = (lane == 0) ? (lab - l0) : NEGV;   // s=1
  float A2 = NEGV;                              // s=64

  // Prefetch t = 1 emits.
  float pb = actb[tstride];
  float ps = actb[tstride + e1];
  float pl = lse[B_DIM + b];

  for (int t = 1; t < Tb; ++t) {
    const float cb = pb, cs_ = ps, cl = pl;
    if (t + 1 < Tb) {                       // software-pipeline next gathers
      const size_t base = (size_t)(t + 1) * tstride;
      pb = actb[base];
      ps = actb[base + e1];
      pl = lse[(t + 1) * B_DIM + b];
    }
    const float eBlank = cb - cl;           // logp of blank
    const float eLab   = cs_ - cl;          // logp of this lane's label

    float upA1 = __shfl_up(A1, 1, 32);      // alpha[2l-1]
    if (lane == 0) upA1 = NEGV;

    // even state s=2l: from s, s-1 (blank cannot skip: ext==0)
    float n0 = lae2(A0, upA1) + eBlank;
    // odd state s=2l+1: from s, s-1, optionally s-2
    float n1 = lae3(A1, A0, cs1 ? upA1 : NEGV) + eLab;
    // state 64 (lane 31): blank, from s=64 and s=63 (own regs)
    float n64 = lae2(A2, A1) + eBlank;

    A0 = valid0 ? n0 : NEGV;
    A1 = valid1 ? n1 : NEGV;
    A2 = (valid64 && lane == 31) ? n64 : NEGV;
  }

  // loss_b = -logaddexp(alpha[2L], alpha[2L-1])
  const float aEnd  = (L < 32) ? __shfl(A0, L, 32) : __shfl(A2, 31, 32);
  const float aEnd1 = __shfl(A1, L - 1, 32);
  if (lane == 0) loss_out[b] = -lae2(aEnd, aEnd1);
}

// ---------------------------------------------------------------------------
// Kernel 3: sum the 32 per-example losses into d_out[0].
// Uses V_WMMA_F32_16X16X4_F32: A = losses (K=0 / K=2 slots per the 32-bit
// 16x4 A layout), B = ones  =>  D[i][j] = v_i + v_{i+16}.  Summing one
// lane's 8 D-VGPRs and adding shfl_xor(16) yields the exact 32-value sum.
// ---------------------------------------------------------------------------
__global__ void __launch_bounds__(32) finalize_kernel(const float* __restrict__ lossv,
                                                      float* __restrict__ out) {
  const int lane = threadIdx.x;           // EXEC all-ones: 32 threads, no branch
  const float v = lossv[lane];
#if __has_builtin(__builtin_amdgcn_wmma_f32_16x16x4_f32)
  v2f a;  a[0] = v;    a[1] = 0.0f;       // A[M][0]=v_M (lanes 0-15), A[M][2]=v_{M+16}
  v2f bb; bb[0] = 1.0f; bb[1] = 1.0f;     // B = ones
  v8f c = {0.f, 0.f, 0.f, 0.f, 0.f, 0.f, 0.f, 0.f};
  v8f d = __builtin_amdgcn_wmma_f32_16x16x4_f32(
      /*neg_a=*/false, a, /*neg_b=*/false, bb,
      /*c_mod=*/(short)0, c, /*reuse_a=*/false, /*reuse_b=*/false);
  float p = d[0] + d[1] + d[2] + d[3] + d[4] + d[5] + d[6] + d[7];
  p += __shfl_xor(p, 16, 32);             // rows 0-7 half + rows 8-15 half
  if (lane == 0) out[0] = p;
#else
  float p = v;
  #pragma unroll
  for (int m = 16; m; m >>= 1) p += __shfl_xor(p, m, 32);
  if (lane == 0) out[0] = p;
#endif
}

// ---------------------------------------------------------------------------
extern "C" void kernel_launch(void* const* d_in, const int* in_sizes, int n_in,
                              void* d_out, int out_size, void* d_ws, size_t ws_size,
                              hipStream_t stream) {
  const float* acts       = (const float*)d_in[0];   // (T,B,V) f32
  const int*   labels     = (const int*)d_in[1];     // flat concat
  const int*   act_lens   = (const int*)d_in[2];     // (B,)
  const int*   label_lens = (const int*)d_in[3];     // (B,)
  float* out = (float*)d_out;

  float* lse   = (float*)d_ws;                        // T*B floats (64 KB)
  float* lossv = lse + (size_t)T_DIM * B_DIM;         // B floats

  lse_kernel<<<T_DIM * B_DIM, 32, 0, stream>>>(acts, lse);
  ctc_alpha_kernel<<<B_DIM, 32, 0, stream>>>(acts, labels, act_lens, label_lens,
                                             lse, lossv);
  finalize_kernel<<<1, 32, 0, stream>>>(lossv, out);
}